// VQ_CVAE_40810779246798
// MI455X (gfx1250) — compile-verified
//
#include <hip/hip_runtime.h>
#include <hip/hip_bf16.h>

// ---------------------------------------------------------------------------
// VQ-CVAE forward on MI455X (gfx1250, wave32).
// Three GEMMs on V_WMMA_F32_16X16X4_F32 (native fp32 matrix op), 4 column
// tiles per block (one A-fragment feeds 4 WMMAs -> 4x less A traffic).
// NCHW outputs double as the A operand of the next GEMM (coalesced reads),
// so no row-major workspace copies are needed.
// ---------------------------------------------------------------------------

typedef float v2f __attribute__((ext_vector_type(2)));
typedef float v8f __attribute__((ext_vector_type(8)));

__device__ __forceinline__ v8f wmma_f32_16x16x4(v2f a, v2f b, v8f c) {
    // 8 args: (neg_a, A, neg_b, B, c_mod, C, reuse_a, reuse_b)
    return __builtin_amdgcn_wmma_f32_16x16x4_f32(false, a, false, b,
                                                 (short)0, c, false, false);
}

// ----- async global->LDS staging (gfx1250), guarded for portability --------
#ifndef __has_builtin
#define __has_builtin(x) 0
#endif
#if __has_builtin(__builtin_amdgcn_global_load_async_to_lds_b32)
#define HAVE_ASYNC_LDS 1
typedef __attribute__((address_space(1))) int* gas_i32p;
typedef __attribute__((address_space(3))) int* las_i32p;
#endif

__device__ __forceinline__ void stage_b32(float* lds_dst, const float* gsrc) {
#ifdef HAVE_ASYNC_LDS
    __builtin_amdgcn_global_load_async_to_lds_b32(
        (gas_i32p)(uintptr_t)gsrc,
        (las_i32p)(uint32_t)(uintptr_t)lds_dst,
        0, 0);
#else
    *lds_dst = *gsrc;
#endif
}

__device__ __forceinline__ void stage_wait() {
#ifdef HAVE_ASYNC_LDS
#if __has_builtin(__builtin_amdgcn_s_wait_asynccnt)
    __builtin_amdgcn_s_wait_asynccnt(0);
#else
    asm volatile("s_wait_asynccnt 0x0" ::: "memory");
#endif
#endif
    __syncthreads();
}

// Sizes
#define NROWS 32768   // 32 batches * 32 * 32 patch positions
#define DCH   512     // latent channels
#define KCB   512     // codebook size
#define KE    192     // 3 * 8 * 8 patch elems
#define CT    4       // column tiles processed per block (64 columns)

// -------------------------------------------------------------------------
// en[k] = sum_d emb_w[d,k]^2   (coalesced over k)
// -------------------------------------------------------------------------
__global__ void k_enorm(const float* __restrict__ emb_w, float* __restrict__ en) {
    int k = blockIdx.x * 256 + threadIdx.x;
    float s = 0.f;
    for (int d = 0; d < DCH; ++d) {
        float v = emb_w[(size_t)d * KCB + k];
        s += v * v;
    }
    en[k] = s;
}

// -------------------------------------------------------------------------
// Encoder: z_e[n, m] = sum_k patch(n)[k] * enc_w[m, k] + enc_b[m]
// grid = (512/64, 32768/128), 256 thr = 8 waves, wave tile 16 x 64.
// B tiles (192 x 16 each, pad 196) in LDS; one A frag -> 4 WMMAs.
// D rows land on consecutive NCHW addresses -> 2x b128 store per tile.
// -------------------------------------------------------------------------
__global__ __launch_bounds__(256) void k_encode(
    const float* __restrict__ x, const float* __restrict__ enc_w,
    const float* __restrict__ enc_b, float* __restrict__ z_e)
{
    __shared__ float Bs[CT * 16 * 196];                 // 50 KB
    const int col0 = blockIdx.x * (CT * 16);
    const int row0 = blockIdx.y * 128;
    const int tid = threadIdx.x;

    for (int e = tid; e < CT * 16 * KE; e += 256) {
        int t = e / (16 * KE), r = e - t * (16 * KE);
        int c = r / KE, k = r - c * KE;
        stage_b32(&Bs[t * 3136 + c * 196 + k],
                  &enc_w[(size_t)(col0 + t * 16 + c) * KE + k]);
    }
    stage_wait();

    const int wave = tid >> 5, lane = tid & 31;
    const int lhalf = lane >> 4, l16 = lane & 15;
    const int k0 = lhalf * 2;                           // ISA A/B frag K pair
    const int nrow = row0 + wave * 16 + l16;            // A row for this lane
    const int b = nrow >> 10, rem = nrow & 1023;
    const int h = rem >> 5, w = rem & 31;
    const float* xbase = x + (size_t)b * 196608 + (size_t)(h * 8) * 256 + w * 8;

    v8f acc[CT] = {};
    #pragma unroll 4
    for (int kk = 0; kk < KE; kk += 4) {
        int k = kk + k0;                                // even -> float2 in-row
        int c = k >> 6, wi = k & 63;
        int ph = wi >> 3, pw = wi & 7;
        v2f a = *(const v2f*)(xbase + (size_t)c * 65536 + ph * 256 + pw);
        #pragma unroll
        for (int t = 0; t < CT; ++t) {
            v2f bf = *(const v2f*)(&Bs[t * 3136 + l16 * 196 + k0 + kk]);
            acc[t] = wmma_f32_16x16x4(a, bf, acc[t]);
        }
    }

    const int nbase = row0 + wave * 16 + 8 * lhalf;     // rows nbase..nbase+7
    const int bb = nbase >> 10, rr = nbase & 1023;
    #pragma unroll
    for (int t = 0; t < CT; ++t) {
        const int m = col0 + t * 16 + l16;
        const float bias = enc_b[m];
        float* dst = z_e + (size_t)bb * 524288 + (size_t)m * 1024 + rr;
        float4 lo = make_float4(acc[t][0] + bias, acc[t][1] + bias,
                                acc[t][2] + bias, acc[t][3] + bias);
        float4 hi = make_float4(acc[t][4] + bias, acc[t][5] + bias,
                                acc[t][6] + bias, acc[t][7] + bias);
        *(float4*)dst = lo;
        *(float4*)(dst + 4) = hi;
    }
}

// -------------------------------------------------------------------------
// Assignment: S = Z x W, fused argmin_k( en[k] - 2*S[n,k] ).
// A read straight from NCHW z_e (coalesced b32 across lanes).
// 8 passes of 4 column tiles; W tiles (512 x 16, pad 516) in LDS (132 KB).
// -------------------------------------------------------------------------
__global__ __launch_bounds__(256) void k_assign(
    const float* __restrict__ z_e, const float* __restrict__ emb_w,
    const float* __restrict__ en,
    int* __restrict__ idx_ws, float* __restrict__ argmin_out)
{
    __shared__ float Bs[CT * 16 * 516];
    const int row0 = blockIdx.x * 128;
    const int tid = threadIdx.x;
    const int wave = tid >> 5, lane = tid & 31;
    const int lhalf = lane >> 4, l16 = lane & 15;
    const int k0 = lhalf * 2;
    const int nrow = row0 + wave * 16 + l16;
    const int bb = nrow >> 10, rr = nrow & 1023;
    const float* abase = z_e + (size_t)bb * 524288 + rr;   // + d*1024

    float bestv[8]; int besti[8];
    #pragma unroll
    for (int v = 0; v < 8; ++v) { bestv[v] = 3.4e38f; besti[v] = 0; }

    for (int cp = 0; cp < KCB / (CT * 16); ++cp) {         // 8 passes
        const int col0 = cp * (CT * 16);
        __builtin_prefetch(emb_w + col0, 0, 0);            // global_prefetch_b8
        __syncthreads();
        for (int e = tid; e < CT * 16 * DCH; e += 256) {
            int t = e >> 13, r = e & 8191;
            int c = r >> 9, d = r & 511;
            stage_b32(&Bs[t * 8256 + c * 516 + d],
                      &emb_w[(size_t)d * KCB + col0 + t * 16 + c]);
        }
        stage_wait();

        v8f acc[CT] = {};
        #pragma unroll 2
        for (int kk = 0; kk < DCH; kk += 4) {
            int d = kk + k0;
            v2f a;
            a.x = abase[(size_t)d * 1024];
            a.y = abase[(size_t)(d + 1) * 1024];
            #pragma unroll
            for (int t = 0; t < CT; ++t) {
                v2f bf = *(const v2f*)(&Bs[t * 8256 + l16 * 516 + k0 + kk]);
                acc[t] = wmma_f32_16x16x4(a, bf, acc[t]);
            }
        }

        #pragma unroll
        for (int t = 0; t < CT; ++t) {
            const float ek = en[col0 + t * 16 + l16];
            #pragma unroll
            for (int v = 0; v < 8; ++v) {
                float val = ek - 2.f * acc[t][v];
                int id = col0 + t * 16 + l16;
                #pragma unroll
                for (int off = 1; off < 16; off <<= 1) {   // min over 16 cols
                    float ov = __shfl_xor(val, off, 32);
                    int oi = __shfl_xor(id, off, 32);
                    if (ov < val || (ov == val && oi < id)) { val = ov; id = oi; }
                }
                if (val < bestv[v]) { bestv[v] = val; besti[v] = id; }
            }
        }
    }

    if (l16 == 0) {
        #pragma unroll
        for (int v = 0; v < 8; ++v) {
            int n = row0 + wave * 16 + v + 8 * lhalf;
            idx_ws[n] = besti[v];
            argmin_out[n] = (float)besti[v];
        }
    }
}

// -------------------------------------------------------------------------
// Gather: emb[n,d] = emb_w[d, idx[n]] written in NCHW layout (this buffer is
// also the decoder's A operand).
// -------------------------------------------------------------------------
__global__ void k_gather(const float* __restrict__ emb_w,
                         const int* __restrict__ idx_ws,
                         float* __restrict__ emb_out)
{
    const int n = blockIdx.x;
    const int id = idx_ws[n];
    const int b = n >> 10, rem = n & 1023;
    for (int d = threadIdx.x; d < DCH; d += 256) {
        emb_out[(size_t)b * 524288 + (size_t)d * 1024 + rem] =
            emb_w[(size_t)d * KCB + id];
    }
}

// -------------------------------------------------------------------------
// Decoder: R[n, j] = sum_d q[n,d] * dec_w[d, j] + dec_b[o(j)], scattered to
// recon NCHW (each latent position owns one 8x8 output patch).
// grid = (192/64, 256). A read straight from NCHW emb buffer.
// -------------------------------------------------------------------------
__global__ __launch_bounds__(256) void k_decode(
    const float* __restrict__ q_nchw, const float* __restrict__ dec_w,
    const float* __restrict__ dec_b, float* __restrict__ recon)
{
    __shared__ float Bs[CT * 16 * 516];
    const int col0 = blockIdx.x * (CT * 16);
    const int row0 = blockIdx.y * 128;
    const int tid = threadIdx.x;

    for (int e = tid; e < CT * 16 * DCH; e += 256) {
        int t = e >> 13, r = e & 8191;
        int c = r >> 9, d = r & 511;
        stage_b32(&Bs[t * 8256 + c * 516 + d],
                  &dec_w[(size_t)d * KE + col0 + t * 16 + c]);
    }
    stage_wait();

    const int wave = tid >> 5, lane = tid & 31;
    const int lhalf = lane >> 4, l16 = lane & 15;
    const int k0 = lhalf * 2;
    const int nrow = row0 + wave * 16 + l16;
    const int bb = nrow >> 10, rr = nrow & 1023;
    const float* abase = q_nchw + (size_t)bb * 524288 + rr;

    v8f acc[CT] = {};
    #pragma unroll 2
    for (int kk = 0; kk < DCH; kk += 4) {
        int d = kk + k0;
        v2f a;
        a.x = abase[(size_t)d * 1024];
        a.y = abase[(size_t)(d + 1) * 1024];
        #pragma unroll
        for (int t = 0; t < CT; ++t) {
            v2f bf = *(const v2f*)(&Bs[t * 8256 + l16 * 516 + k0 + kk]);
            acc[t] = wmma_f32_16x16x4(a, bf, acc[t]);
        }
    }

    #pragma unroll
    for (int t = 0; t < CT; ++t) {
        const int j = col0 + t * 16 + l16;
        const int o = j >> 6, wi = j & 63, ph = wi >> 3, pw = wi & 7;
        const float bias = dec_b[o];
        #pragma unroll
        for (int v = 0; v < 8; ++v) {
            int n = row0 + wave * 16 + v + 8 * lhalf;
            int b = n >> 10, h = (n >> 5) & 31, wp = n & 31;
            recon[(size_t)b * 196608 + (size_t)o * 65536 +
                  (size_t)(h * 8 + ph) * 256 + wp * 8 + pw] = acc[t][v] + bias;
        }
    }
}

// ---------------------------------------------------------------------------
extern "C" void kernel_launch(void* const* d_in, const int* in_sizes, int n_in,
                              void* d_out, int out_size, void* d_ws, size_t ws_size,
                              hipStream_t stream) {
    const float* x     = (const float*)d_in[0];   // (32,3,256,256)
    const float* enc_w = (const float*)d_in[1];   // (512,3,8,8)
    const float* enc_b = (const float*)d_in[2];   // (512,)
    const float* dec_w = (const float*)d_in[3];   // (512,3,8,8)
    const float* dec_b = (const float*)d_in[4];   // (3,)
    const float* emb_w = (const float*)d_in[5];   // (512,512)

    float* out        = (float*)d_out;
    float* recon      = out;                       // 6,291,456
    float* z_e_out    = out + 6291456;             // 16,777,216
    float* emb_out    = out + 23068672;            // 16,777,216
    float* argmin_out = out + 39845888;            // 32,768

    int*   idx_ws = (int*)d_ws;                    // N i32
    float* en_ws  = (float*)(idx_ws + NROWS);      // 512 f32

    k_enorm<<<KCB / 256, 256, 0, stream>>>(emb_w, en_ws);
    k_encode<<<dim3(DCH / (CT * 16), NROWS / 128), 256, 0, stream>>>(
        x, enc_w, enc_b, z_e_out);
    k_assign<<<NROWS / 128, 256, 0, stream>>>(
        z_e_out, emb_w, en_ws, idx_ws, argmin_out);
    k_gather<<<NROWS, 256, 0, stream>>>(emb_w, idx_ws, emb_out);
    k_decode<<<dim3(KE / (CT * 16), NROWS / 128), 256, 0, stream>>>(
        emb_out, dec_w, dec_b, recon);
}